// GCN_13821204758566
// MI455X (gfx1250) — compile-verified
//
#include <hip/hip_runtime.h>
#include <cstdint>

#define B_GRAPHS 8
#define N_NODES  10000
#define N_EDGES  320000
#define HDIM     128   // F == H == 128

typedef float v2f __attribute__((ext_vector_type(2)));
typedef float v8f __attribute__((ext_vector_type(8)));

// ---------------------------------------------------------------------------
// Small elementwise helpers
// ---------------------------------------------------------------------------
__global__ __launch_bounds__(256) void init_deg_kernel(float* deg, int n) {
    int i = blockIdx.x * 256 + threadIdx.x;
    if (i < n) deg[i] = 1.0f;                       // self-loop contributes 1
}

__global__ __launch_bounds__(256) void count_deg_kernel(const int* __restrict__ ei,
                                                        float* __restrict__ deg) {
    int i = blockIdx.x * 256 + threadIdx.x;
    if (i >= B_GRAPHS * N_EDGES) return;
    int g = i / N_EDGES, e = i - g * N_EDGES;
    int col = ei[(g * 2 + 1) * N_EDGES + e];
    atomicAdd(&deg[g * N_NODES + col], 1.0f);
}

__global__ __launch_bounds__(256) void rsqrt_kernel(float* d, int n) {
    int i = blockIdx.x * 256 + threadIdx.x;
    if (i < n) d[i] = rsqrtf(d[i]);                 // deg >= 1 always
}

__global__ __launch_bounds__(256) void zero_kernel(float* p, int n) {
    int i = blockIdx.x * 256 + threadIdx.x;
    if (i < n) p[i] = 0.0f;
}

// ---------------------------------------------------------------------------
// GEMM: Y[rows x 128] = X[rows x 128] @ W[128 x 128], per graph (blockIdx.y).
// 256 threads = 8 waves. Block stages a 128x128 X tile in LDS via
// GLOBAL_LOAD_ASYNC_TO_LDS_B128 (ASYNCcnt path, swizzled destinations),
// wave w computes rows [w*16, w*16+16) for all 8 N-tiles with f32 WMMA.
// ---------------------------------------------------------------------------
__global__ __launch_bounds__(256) void gemm128_wmma(const float* __restrict__ X,
                                                    const float* __restrict__ W,
                                                    float* __restrict__ Y,
                                                    int nrows) {
    __shared__ float As[128 * 128];                 // 64 KB, swizzled rows

    const int g    = blockIdx.y;
    const int row0 = blockIdx.x * 128;
    const float* Xg = X + (size_t)g * N_NODES * HDIM;
    float*       Yg = Y + (size_t)g * N_NODES * HDIM;

    // ---- stage X tile: 4096 x b128 async copies straight into LDS ----
    const int t = threadIdx.x;
#pragma unroll
    for (int i = 0; i < 16; ++i) {
        int f4   = t + i * 256;                     // 0..4095
        int r    = f4 >> 5;                         // row in tile
        int k    = (f4 & 31) << 2;                  // col, multiple of 4
        int grow = row0 + r;
        int ks   = (k + (r << 2)) & 127;            // swizzle: conflict-free A reads
        float* lp = As + r * 128 + ks;
        if (grow < nrows) {
            unsigned lds_off = (unsigned)(uintptr_t)lp;
            const float* gp  = Xg + (size_t)grow * HDIM + k;
            asm volatile("global_load_async_to_lds_b128 %0, %1, off"
                         :: "v"(lds_off), "v"(gp)
                         : "memory");
        } else {
            *(float4*)lp = make_float4(0.f, 0.f, 0.f, 0.f);   // zero tail rows
        }
    }
    asm volatile("s_wait_asynccnt 0x0" ::: "memory");
    __syncthreads();

    const int wave = t >> 5;
    const int lane = t & 31;
    const int lm   = lane & 15;
    const int hf   = lane >> 4;                     // lane half (0/1)
    const int mrow = wave * 16 + lm;                // A-frag row in tile
    const int msw  = mrow << 2;                     // row swizzle shift

    v8f c[8] = {};                                  // 8 N-tile accumulators

    for (int kk = 0; kk < 32; ++kk) {
        const int k0 = kk * 4 + hf * 2;             // K=0/1 (lanes 0-15), K=2/3 (16-31)
        v2f a;
        a.x = As[mrow * 128 + ((k0     + msw) & 127)];
        a.y = As[mrow * 128 + ((k0 + 1 + msw) & 127)];
#pragma unroll
        for (int nt = 0; nt < 8; ++nt) {
            v2f b;
            b.x = W[(k0)     * 128 + nt * 16 + lm];
            b.y = W[(k0 + 1) * 128 + nt * 16 + lm];
            // (neg_a, A, neg_b, B, c_mod, C, reuse_a, reuse_b)
            c[nt] = __builtin_amdgcn_wmma_f32_16x16x4_f32(
                false, a, false, b, (short)0, c[nt], false, false);
        }
    }

    // ---- store D: VGPR r -> rows (r, r+8) of the wave's 16-row strip ----
#pragma unroll
    for (int nt = 0; nt < 8; ++nt) {
#pragma unroll
        for (int r = 0; r < 8; ++r) {
            int row = row0 + wave * 16 + hf * 8 + r;
            if (row < nrows) Yg[(size_t)row * HDIM + nt * 16 + lm] = c[nt][r];
        }
    }
}

// ---------------------------------------------------------------------------
// Edge scatter: one wave per edge; lane handles features lane, lane+32, ...
// dst[col] += src[row] * dinv[row] * dinv[col]   (L2-resident atomics)
// ---------------------------------------------------------------------------
__global__ __launch_bounds__(256) void scatter_kernel(const int* __restrict__ ei,
                                                      const float* __restrict__ dinv,
                                                      const float* __restrict__ src,
                                                      float* __restrict__ dst) {
    int tid  = blockIdx.x * 256 + threadIdx.x;
    int w    = tid >> 5;
    int lane = tid & 31;
    if (w >= B_GRAPHS * N_EDGES) return;
    int g = w / N_EDGES, e = w - g * N_EDGES;
    int row = ei[(g * 2 + 0) * N_EDGES + e];
    int col = ei[(g * 2 + 1) * N_EDGES + e];
    float nrm = dinv[g * N_NODES + row] * dinv[g * N_NODES + col];
    const float* s = src + ((size_t)g * N_NODES + row) * HDIM;
    float*       d = dst + ((size_t)g * N_NODES + col) * HDIM;
#pragma unroll
    for (int j = 0; j < 4; ++j) {
        int f = lane + j * 32;
        atomicAdd(d + f, s[f] * nrm);
    }
}

// ---------------------------------------------------------------------------
// Threefry-2x32 (JAX-compatible), key = PRNGKey(42) = (0, 42).
// Reproduces jax.random.bernoulli(key(42), 0.5, (N, H)).
// ---------------------------------------------------------------------------
__device__ __forceinline__ unsigned rotl32(unsigned x, int r) {
    return (x << r) | (x >> (32 - r));
}

__device__ unsigned threefry_bits(unsigned idx) {
    const unsigned half = (unsigned)(N_NODES * HDIM) / 2u;   // 640000
    unsigned i  = (idx < half) ? idx : idx - half;
    unsigned x0 = i, x1 = i + half;
    const unsigned k0 = 0u, k1 = 42u, k2 = k0 ^ k1 ^ 0x1BD11BDAu;
    x0 += k0; x1 += k1;
#define TF_ROUND(r) { x0 += x1; x1 = rotl32(x1, (r)); x1 ^= x0; }
#define TF_G0 TF_ROUND(13) TF_ROUND(15) TF_ROUND(26) TF_ROUND(6)
#define TF_G1 TF_ROUND(17) TF_ROUND(29) TF_ROUND(16) TF_ROUND(24)
    TF_G0 x0 += k1; x1 += k2 + 1u;
    TF_G1 x0 += k2; x1 += k0 + 2u;
    TF_G0 x0 += k0; x1 += k1 + 3u;
    TF_G1 x0 += k1; x1 += k2 + 4u;
    TF_G0 x0 += k2; x1 += k0 + 5u;
#undef TF_G1
#undef TF_G0
#undef TF_ROUND
    return (idx < half) ? x0 : x1;
}

// ---------------------------------------------------------------------------
// Finalize layer 1: h = dropout(relu(agg + selfloop + b1))
// ---------------------------------------------------------------------------
__global__ __launch_bounds__(256) void finalize1_kernel(const float* __restrict__ xw,
                                                        const float* __restrict__ dinv,
                                                        const float* __restrict__ b1,
                                                        float* __restrict__ h) {
    const int NH  = N_NODES * HDIM;
    int idx = blockIdx.x * 256 + threadIdx.x;
    if (idx >= B_GRAPHS * NH) return;
    int g = idx / NH, rem = idx - g * NH;
    int n = rem >> 7, f = rem & 127;
    float di  = dinv[g * N_NODES + n];
    float acc = h[idx] + xw[idx] * di * di + b1[f];
    acc = fmaxf(acc, 0.0f);
    unsigned bits = threefry_bits((unsigned)rem);
    float u = __uint_as_float((bits >> 9) | 0x3f800000u) - 1.0f;
    h[idx] = (u < 0.5f) ? acc * 2.0f : 0.0f;        // /(1-p) with p=0.5
}

// Finalize layer 2: out += selfloop + b2
__global__ __launch_bounds__(256) void finalize2_kernel(const float* __restrict__ xw,
                                                        const float* __restrict__ dinv,
                                                        const float* __restrict__ b2,
                                                        float* __restrict__ out) {
    const int NH  = N_NODES * HDIM;
    int idx = blockIdx.x * 256 + threadIdx.x;
    if (idx >= B_GRAPHS * NH) return;
    int g = idx / NH, rem = idx - g * NH;
    int n = rem >> 7, f = rem & 127;
    float di = dinv[g * N_NODES + n];
    out[idx] = out[idx] + xw[idx] * di * di + b2[f];
}

// ---------------------------------------------------------------------------
// Launch
// ---------------------------------------------------------------------------
extern "C" void kernel_launch(void* const* d_in, const int* in_sizes, int n_in,
                              void* d_out, int out_size, void* d_ws, size_t ws_size,
                              hipStream_t stream) {
    const float* xs = (const float*)d_in[0];
    const int*   ei = (const int*)  d_in[1];
    const float* W1 = (const float*)d_in[2];
    const float* b1 = (const float*)d_in[3];
    const float* W2 = (const float*)d_in[4];
    const float* b2 = (const float*)d_in[5];
    float* out = (float*)d_out;

    char* ws = (char*)d_ws;
    float* dinv = (float*)ws;                                   // B*N
    size_t off  = (((size_t)B_GRAPHS * N_NODES * 4) + 255) & ~(size_t)255;
    float* xw   = (float*)(ws + off);                           // B*N*H (reused for both layers)
    size_t fsz  = (size_t)B_GRAPHS * N_NODES * HDIM * 4;
    float* h    = (float*)(ws + off + fsz);                     // B*N*H

    const int bn  = B_GRAPHS * N_NODES;
    const int be  = B_GRAPHS * N_EDGES;
    const int bnh = B_GRAPHS * N_NODES * HDIM;

    // degrees -> dinv
    init_deg_kernel <<<(bn + 255) / 256, 256, 0, stream>>>(dinv, bn);
    count_deg_kernel<<<(be + 255) / 256, 256, 0, stream>>>(ei, dinv);
    rsqrt_kernel    <<<(bn + 255) / 256, 256, 0, stream>>>(dinv, bn);

    dim3 ggrid((N_NODES + 127) / 128, B_GRAPHS);

    // layer 1: xw = x @ W1 ; h = dropout(relu(agg(xw) + b1))
    gemm128_wmma<<<ggrid, 256, 0, stream>>>(xs, W1, xw, N_NODES);
    zero_kernel <<<(bnh + 255) / 256, 256, 0, stream>>>(h, bnh);
    scatter_kernel  <<<(be + 7) / 8, 256, 0, stream>>>(ei, dinv, xw, h);
    finalize1_kernel<<<(bnh + 255) / 256, 256, 0, stream>>>(xw, dinv, b1, h);

    // layer 2: xw = h @ W2 ; out = agg(xw) + b2
    gemm128_wmma<<<ggrid, 256, 0, stream>>>(h, W2, xw, N_NODES);
    zero_kernel <<<(bnh + 255) / 256, 256, 0, stream>>>(out, bnh);
    scatter_kernel  <<<(be + 7) / 8, 256, 0, stream>>>(ei, dinv, xw, out);
    finalize2_kernel<<<(bnh + 255) / 256, 256, 0, stream>>>(xw, dinv, b2, out);
}